// PredictionHead_74526272520964
// MI455X (gfx1250) — compile-verified
//
#include <hip/hip_runtime.h>
#include <hip/hip_bf16.h>
#include <stddef.h>

// ---------------------------------------------------------------------------
// PredictionHead for gfx1250 (MI455X):
//   z = x @ W^T + b  (only the 23 patches that reach the 720-step horizon),
//   overlap-add fold (stride 32, patch 64), keep last 720 columns, / count.
// GEMM runs on V_WMMA_F32_16X16X4_F32 (full fp32 precision matrix pipe).
// ---------------------------------------------------------------------------

typedef __attribute__((ext_vector_type(2))) float v2f;
typedef __attribute__((ext_vector_type(8))) float v8f;

#define D_MODEL   1024
#define PATCH     64
#define NPATCH    256
#define BATCH     512
#define STRIDE    32
#define PRED      720
#define TOTAL_LEN 8224          // (256-1)*32 + 64
#define T0        (TOTAL_LEN - PRED)   // 7504
#define I_FIRST   233           // first patch index that touches t >= T0
#define N_I       23            // patches 233..255
#define M_ROWS    (BATCH * N_I) // 11776 GEMM rows
#define KBLK      128           // K-slab staged in LDS
#define LDS_STR   (KBLK + 4)    // 132 floats: row stride padding kills bank conflicts

// ---------------------------------------------------------------------------
// Kernel 1: z[M_ROWS, 64] = x_rows @ W^T  via f32 WMMA.
// Wave tile: 16 (M) x 64 (N). 8 waves / block -> 92 blocks (736 M-tiles).
//
// f32 16x16x4 fragment layouts (wave32):
//   A (16x4):  lane l: M = l%16;  K = {k, k+1} (lanes 0-15), {k+2, k+3} (16-31)
//   B (4x16):  lane l: N = l%16;  K = {k, k+1} (lanes 0-15), {k+2, k+3} (16-31)
//   C/D:       VGPR v, lane l: M = v + 8*(l/16), N = l%16
// ---------------------------------------------------------------------------
__global__ __launch_bounds__(256) void ph_gemm(const float* __restrict__ x,
                                               const float* __restrict__ Wm,
                                               float* __restrict__ z) {
    __shared__ float wlds[PATCH * LDS_STR];   // 64 x 132 floats = 33 KB

    const int tid  = threadIdx.x;
    const int wave = tid >> 5;
    const int lane = tid & 31;
    const int li   = lane & 15;
    const int h    = lane >> 4;

    const int tile_m = blockIdx.x * 8 + wave;          // 0..735
    const int row    = tile_m * 16 + li;               // GEMM row this lane loads A for
    const int bb     = row / N_I;                      // batch
    const int ip     = row - bb * N_I;                 // 0..22 -> patch I_FIRST+ip
    const float* xrow = x + (size_t)(bb * NPATCH + (I_FIRST + ip)) * D_MODEL;

    v8f c0 = {}, c1 = {}, c2 = {}, c3 = {};

    const float4* W4 = (const float4*)Wm;              // W is [64, 1024], 16B aligned
    float4*       L4 = (float4*)wlds;

    for (int k0 = 0; k0 < D_MODEL; k0 += KBLK) {
        __syncthreads();
        // Cooperative stage of W[:, k0:k0+128] into LDS. 2048 float4 / 256 thr.
#pragma unroll
        for (int j = 0; j < 8; ++j) {
            int f   = tid + 256 * j;          // 0..2047
            int n   = f >> 5;                 // 0..63
            int kk4 = f & 31;                 // 0..31
            L4[n * (LDS_STR >> 2) + kk4] = W4[n * (D_MODEL >> 2) + (k0 >> 2) + kk4];
        }
        __syncthreads();

        if (k0 + KBLK < D_MODEL)
            __builtin_prefetch(xrow + k0 + KBLK, 0, 1);   // global_prefetch next A slab

#pragma unroll 4
        for (int kk = 0; kk < KBLK; kk += 4) {
            v2f a  = *(const v2f*)(xrow + k0 + kk + 2 * h);
            v2f b0 = *(const v2f*)(wlds + ( 0 + li) * LDS_STR + kk + 2 * h);
            v2f b1 = *(const v2f*)(wlds + (16 + li) * LDS_STR + kk + 2 * h);
            v2f b2 = *(const v2f*)(wlds + (32 + li) * LDS_STR + kk + 2 * h);
            v2f b3 = *(const v2f*)(wlds + (48 + li) * LDS_STR + kk + 2 * h);
            c0 = __builtin_amdgcn_wmma_f32_16x16x4_f32(false, a, false, b0, (short)0, c0, false, false);
            c1 = __builtin_amdgcn_wmma_f32_16x16x4_f32(false, a, false, b1, (short)0, c1, false, false);
            c2 = __builtin_amdgcn_wmma_f32_16x16x4_f32(false, a, false, b2, (short)0, c2, false, false);
            c3 = __builtin_amdgcn_wmma_f32_16x16x4_f32(false, a, false, b3, (short)0, c3, false, false);
        }
    }

    // Store D: lane l writes M = v + 8*h, N = li (+16 per N-tile)
    const int mbase = tile_m * 16 + h * 8;
#pragma unroll
    for (int v = 0; v < 8; ++v) {
        float* zr = z + (size_t)(mbase + v) * PATCH + li;
        zr[0]  = c0[v];
        zr[16] = c1[v];
        zr[32] = c2[v];
        zr[48] = c3[v];
    }
}

// ---------------------------------------------------------------------------
// Kernel 2: fold. t = T0 + t' (t in [7504, 8224)), q = t/32 in [234, 256].
// Contributors: patch i=q-1 (always, p = t-32q+32 in [32,64)) and patch i=q
// (only if q <= 255, p = t-32q in [0,32)). count = 2 if q<=255 else 1,
// which matches the reference's global count for this t-range. Bias folded in.
// z row index for (batch b, patch i) is b*23 + (i - 233).
// ---------------------------------------------------------------------------
__global__ __launch_bounds__(256) void ph_fold(const float* __restrict__ z,
                                               const float* __restrict__ bias,
                                               float* __restrict__ out) {
    int idx = blockIdx.x * blockDim.x + threadIdx.x;
    if (idx >= BATCH * PRED) return;
    int bb = idx / PRED;
    int tp = idx - bb * PRED;
    int t  = T0 + tp;
    int q  = t >> 5;          // 234..256
    int p  = t & 31;          // t - 32*q

    const float* zb = z + (size_t)bb * N_I * PATCH;

    float val = zb[(q - 1 - I_FIRST) * PATCH + (p + 32)] + bias[p + 32];
    if (q <= NPATCH - 1) {
        val = (val + zb[(q - I_FIRST) * PATCH + p] + bias[p]) * 0.5f;
    }
    out[idx] = val;
}

extern "C" void kernel_launch(void* const* d_in, const int* in_sizes, int n_in,
                              void* d_out, int out_size, void* d_ws, size_t ws_size,
                              hipStream_t stream) {
    const float* x    = (const float*)d_in[0];   // [512, 256, 1024]
    const float* W    = (const float*)d_in[1];   // [64, 1024]
    const float* bias = (const float*)d_in[2];   // [64]
    float* out = (float*)d_out;                  // [512, 720]
    float* z   = (float*)d_ws;                   // [11776, 64] = 2.9 MB scratch

    // 736 M-tiles of 16 rows, 8 waves (tiles) per block -> 92 blocks
    ph_gemm<<<dim3(M_ROWS / 16 / 8), dim3(256), 0, stream>>>(x, W, z);

    int nOut = BATCH * PRED;
    ph_fold<<<dim3((nOut + 255) / 256), dim3(256), 0, stream>>>(z, bias, out);
}